// SoftmaxSurface_62543313764812
// MI455X (gfx1250) — compile-verified
//
#include <hip/hip_runtime.h>

// SoftmaxSurface for MI455X (gfx1250, wave32).
// Pure streaming kernel: 64 MB read + 128 MB written, HBM-bound (~8.3 us at
// 23.3 TB/s). One 256-thread block (8 wave32s) per input row of 4096 floats;
// 16 floats held in registers per thread so each input element is read from
// global memory exactly once. Output written with non-temporal 128-bit stores.
// Input staged via CDNA5 async global->LDS copies (ASYNCcnt path).

typedef __attribute__((ext_vector_type(4))) float v4f;
typedef __attribute__((ext_vector_type(4))) int   v4i;

// Address-space-qualified pointer types for the CDNA5 async-copy builtin:
// clang's diagnostic showed param0 = AS1 (global) v4i*, param1 = AS3 (LDS) v4i*.
typedef __attribute__((address_space(1))) v4i* gv4i_p;
typedef __attribute__((address_space(3))) v4i* lv4i_p;

#define DCOLS   4096
#define THREADS 256
#define CHUNKS  4          // float4 chunks per thread: 4 * 256 * 4 = 4096 floats
#define NWAVES  (THREADS / 32)

#define LOG2E 1.4426950408889634f

// ---------------- wave32 reductions (gfx1250 is wave32-only) ----------------
__device__ __forceinline__ float wred_max(float v) {
#pragma unroll
  for (int off = 16; off > 0; off >>= 1) v = fmaxf(v, __shfl_xor(v, off, 32));
  return v;
}
__device__ __forceinline__ float wred_min(float v) {
#pragma unroll
  for (int off = 16; off > 0; off >>= 1) v = fminf(v, __shfl_xor(v, off, 32));
  return v;
}
__device__ __forceinline__ float wred_sum(float v) {
#pragma unroll
  for (int off = 16; off > 0; off >>= 1) v += __shfl_xor(v, off, 32);
  return v;
}

// Block reductions: each call has its own 8-float LDS slot array, so a single
// barrier between the per-wave write and the broadcast read is sufficient.
__device__ __forceinline__ float bred_max(float v, float* slot) {
  v = wred_max(v);
  if ((threadIdx.x & 31) == 0) slot[threadIdx.x >> 5] = v;
  __syncthreads();
  float r = slot[0];
#pragma unroll
  for (int w = 1; w < NWAVES; ++w) r = fmaxf(r, slot[w]);
  return r;
}
__device__ __forceinline__ float bred_min(float v, float* slot) {
  v = wred_min(v);
  if ((threadIdx.x & 31) == 0) slot[threadIdx.x >> 5] = v;
  __syncthreads();
  float r = slot[0];
#pragma unroll
  for (int w = 1; w < NWAVES; ++w) r = fminf(r, slot[w]);
  return r;
}
__device__ __forceinline__ float bred_sum(float v, float* slot) {
  v = wred_sum(v);
  if ((threadIdx.x & 31) == 0) slot[threadIdx.x >> 5] = v;
  __syncthreads();
  float r = slot[0];
#pragma unroll
  for (int w = 1; w < NWAVES; ++w) r += slot[w];
  return r;
}

__global__ __launch_bounds__(THREADS) void SoftmaxSurface_kernel(
    const float* __restrict__ in, float* __restrict__ out) {
  __shared__ float red[4][NWAVES];

  const int    t   = threadIdx.x;
  const size_t row = blockIdx.x;

  const float* __restrict__ src  = in  + row * (size_t)DCOLS;
  float* __restrict__       dstA = out + row * (size_t)(2 * DCOLS); // out row 2r
  float* __restrict__       dstB = dstA + DCOLS;                    // out row 2r+1

  float v[4 * CHUNKS];

#if __has_builtin(__builtin_amdgcn_global_load_async_to_lds_b128)
  // ---- CDNA5 async global->LDS staging (tracked by ASYNCcnt) ----
  __shared__ float row_lds[DCOLS];  // 16 KB of the 320 KB/WGP pool
#pragma unroll
  for (int k = 0; k < CHUNKS; ++k) {
    const int idx4 = k * THREADS + t;  // float4 index, fully coalesced
    __builtin_amdgcn_global_load_async_to_lds_b128(
        (gv4i_p)(src + idx4 * 4),
        (lv4i_p)(row_lds + idx4 * 4),
        /*offset=*/0, /*cpol=*/0);
  }
#if __has_builtin(__builtin_amdgcn_s_wait_asynccnt)
  __builtin_amdgcn_s_wait_asynccnt(0);
#else
  asm volatile("s_wait_asynccnt 0" ::: "memory");
#endif
  __syncthreads();  // all waves' async copies have landed in LDS
#pragma unroll
  for (int k = 0; k < CHUNKS; ++k) {
    const int idx4 = k * THREADS + t;
    v4f f = ((const v4f*)row_lds)[idx4];
    v[4 * k + 0] = f.x; v[4 * k + 1] = f.y;
    v[4 * k + 2] = f.z; v[4 * k + 3] = f.w;
  }
#else
  // ---- fallback: direct coalesced 128-bit streaming loads ----
#pragma unroll
  for (int k = 0; k < CHUNKS; ++k) {
    const int idx4 = k * THREADS + t;
    v4f f = __builtin_nontemporal_load((const v4f*)src + idx4);
    v[4 * k + 0] = f.x; v[4 * k + 1] = f.y;
    v[4 * k + 2] = f.z; v[4 * k + 3] = f.w;
  }
#endif

  // ---- phase 1: row max ----
  float mx = -__builtin_inff();
#pragma unroll
  for (int i = 0; i < 4 * CHUNKS; ++i) mx = fmaxf(mx, v[i]);
  mx = bred_max(mx, red[0]);

  // ---- phase 2: e1 = exp(x - m) in-place; sum(e1), min(e1) ----
  float s1 = 0.0f, mn = __builtin_inff();
#pragma unroll
  for (int i = 0; i < 4 * CHUNKS; ++i) {
    const float e = __builtin_amdgcn_exp2f((v[i] - mx) * LOG2E);  // v_exp_f32
    v[i] = e;
    s1 += e;
    mn = fminf(mn, e);
  }
  s1 = bred_sum(s1, red[1]);
  mn = bred_min(mn, red[2]);
  const float inv1 = 1.0f / s1;

  // ---- phase 3: store a = e1/s1 (NT); overwrite regs with e2 = exp(mn - e1) ----
  float s2 = 0.0f;
#pragma unroll
  for (int k = 0; k < CHUNKS; ++k) {
    const int idx4 = k * THREADS + t;
    v4f o;
    o.x = v[4 * k + 0] * inv1; o.y = v[4 * k + 1] * inv1;
    o.z = v[4 * k + 2] * inv1; o.w = v[4 * k + 3] * inv1;
    __builtin_nontemporal_store(o, (v4f*)dstA + idx4);
#pragma unroll
    for (int j = 0; j < 4; ++j) {
      const float e2 = __builtin_amdgcn_exp2f((mn - v[4 * k + j]) * LOG2E);
      v[4 * k + j] = e2;
      s2 += e2;
    }
  }
  s2 = bred_sum(s2, red[3]);
  const float inv2 = 1.0f / s2;

  // ---- phase 4: store b = e2/s2 (NT) ----
#pragma unroll
  for (int k = 0; k < CHUNKS; ++k) {
    const int idx4 = k * THREADS + t;
    v4f o;
    o.x = v[4 * k + 0] * inv2; o.y = v[4 * k + 1] * inv2;
    o.z = v[4 * k + 2] * inv2; o.w = v[4 * k + 3] * inv2;
    __builtin_nontemporal_store(o, (v4f*)dstB + idx4);
  }
}

extern "C" void kernel_launch(void* const* d_in, const int* in_sizes, int n_in,
                              void* d_out, int out_size, void* d_ws, size_t ws_size,
                              hipStream_t stream) {
  (void)n_in; (void)out_size; (void)d_ws; (void)ws_size;
  const float* in = (const float*)d_in[0];
  float* out = (float*)d_out;
  const int rows = in_sizes[0] / DCOLS;  // 32 * 128 = 4096
  SoftmaxSurface_kernel<<<rows, THREADS, 0, stream>>>(in, out);
}